// FWPModel_70153995813285
// MI455X (gfx1250) — compile-verified
//
#include <hip/hip_runtime.h>
#include <hip/hip_bf16.h>

typedef float v2f __attribute__((ext_vector_type(2)));
typedef float v8f __attribute__((ext_vector_type(8)));

#define LD      1282      // wu row length: 1026 slow outs + 256 w0-base cols
#define KDIM    256
#define NTOT    1282
#define NTILES  81        // ceil(1282/16)
#define MTILES  256       // 4096/16
#define SSTEPS  512
#define BATCH   8

// ---------------------------------------------------------------------------
// Kernel 1: wu[m, n] = x[m, :] . W[n, :] (+ slow_b[n] for n<1026)
//   W rows 0..1025  -> slow_W
//   W rows 1026..   -> fast_W0   (precomputes the w0 @ x contribution, no bias)
// One 16x16 output tile per wave, K=256 -> 64 x v_wmma_f32_16x16x4_f32.
// ---------------------------------------------------------------------------
__global__ __launch_bounds__(256)
void fwp_slow_gemm(const float* __restrict__ x,
                   const float* __restrict__ slow_W,
                   const float* __restrict__ slow_b,
                   const float* __restrict__ fast_W0,
                   float* __restrict__ wu)
{
    int wid  = threadIdx.x >> 5;
    int lane = threadIdx.x & 31;
    int wt   = blockIdx.x * 8 + wid;
    if (wt >= MTILES * NTILES) return;
    int mt = wt / NTILES, nt = wt % NTILES;
    int m0 = mt * 16, n0 = nt * 16;
    int q = lane & 15, h = lane >> 4;

    int n = n0 + q;
    // per-lane B (weight) row pointer; clamp out-of-range lanes to valid memory
    const float* wp;
    if (n < 1026) {
        wp = slow_W + (size_t)n * KDIM;
    } else {
        int idx = n - 1026;
        if (idx > 255) idx = 255;           // n>=1282 lanes: garbage-but-safe
        wp = fast_W0 + (size_t)idx * KDIM;
    }
    float bv = (n < 1026) ? slow_b[n] : 0.0f;

    v8f c;
    #pragma unroll
    for (int r = 0; r < 8; ++r) c[r] = bv;   // bias replicated down the M dim

    // A frag: lane q = row m0+q; VGPR0/1 hold K = 2h, 2h+1 of each 4-chunk
    const float* ap = x + (size_t)(m0 + q) * KDIM + 2 * h;
    const float* bp = wp + 2 * h;

    #pragma unroll 8
    for (int k = 0; k < KDIM; k += 4) {
        v2f a = *(const v2f*)(ap + k);
        v2f b = *(const v2f*)(bp + k);
        c = __builtin_amdgcn_wmma_f32_16x16x4_f32(false, a, false, b,
                                                  (short)0, c, false, false);
    }

    if (n < NTOT) {
        #pragma unroll
        for (int r = 0; r < 8; ++r) {
            int m = m0 + h * 8 + r;          // C layout: lanes 0-15 M=r, 16-31 M=8+r
            wu[(size_t)m * LD + n] = c[r];
        }
    }
}

// ---------------------------------------------------------------------------
// Kernel 2: sequential scan. grid = (16 o-blocks, 8 batches), 256 threads.
// Wave w owns i-columns [32w, 32w+32): two 16x16 WMMA tiles. fw slice
// (16 o x 256 i) lives in VGPRs in WMMA C layout (16 regs/lane).
// Per step: add/sub outer products via v_wmma_f32_16x16x4_f32 (K=8 -> 2 WMMA
// per tile), gate-scaled accumulate into fw, then GEMV partials + reduction.
// Next step's 41KB wu window + x row are prefetched (global_prefetch_b8) to
// hide L2 latency behind the current step's WMMA/VALU work.
// ---------------------------------------------------------------------------
__global__ __launch_bounds__(256)
void fwp_scan(const float* __restrict__ wu,
              const float* __restrict__ x,
              float* __restrict__ out)
{
    __shared__ float red[8 * 16];

    int tid  = threadIdx.x;
    int w    = tid >> 5;
    int lane = tid & 31;
    int q    = lane & 15;
    int h    = lane >> 4;
    int o0   = blockIdx.x * 16;   // output-row block
    int b    = blockIdx.y;        // batch
    int c0   = 32 * w + q;        // tile-0 column (i index)
    int c1   = c0 + 16;           // tile-1 column

    v8f fw0 = {0,0,0,0,0,0,0,0};
    v8f fw1 = {0,0,0,0,0,0,0,0};
    const v8f zero = {0,0,0,0,0,0,0,0};

    const int STEP_BYTES = BATCH * LD * 4;   // 8 wu rows per step = 41024 B

    for (int s = 0; s < SSTEPS; ++s) {
        const float* row  = wu + (size_t)(s * BATCH) * LD;  // b'=0 row of step s
        const float* rowb = row + (size_t)b * LD;

        // ---- prefetch step s+1 working set (overlaps with this step) ----
        if (s + 1 < SSTEPS) {
            const char* nb = (const char*)(row) + STEP_BYTES;   // next wu rows
            int off0 = tid * 128;
            int off1 = 32768 + tid * 128;
            if (off0 < STEP_BYTES) __builtin_prefetch(nb + off0, 0, 3);
            if (off1 < STEP_BYTES) __builtin_prefetch(nb + off1, 0, 3);
            if (tid < 64) {                                     // next x row (8KB)
                const char* nx = (const char*)(x + (size_t)(s + 1) * BATCH * 256);
                __builtin_prefetch(nx + tid * 128, 0, 3);
            }
        }

        // gates (uniform per block; computed redundantly per lane)
        float sa = 1.0f / (1.0f + __expf(-rowb[1024]));
        float ss = 1.0f / (1.0f + __expf(-rowb[1025]));

        int kk = 2 * h;   // lane's K rows within a 4-chunk: kk, kk+1

        // A fragments (16 x 4): rows = o0..o0+15 of a1 (col 0..) / s1 (col 512..)
        v2f aA0, aA1, sA0, sA1;
        aA0.x = row[(size_t)(kk + 0) * LD +       o0 + q];
        aA0.y = row[(size_t)(kk + 1) * LD +       o0 + q];
        aA1.x = row[(size_t)(kk + 4) * LD +       o0 + q];
        aA1.y = row[(size_t)(kk + 5) * LD +       o0 + q];
        sA0.x = row[(size_t)(kk + 0) * LD + 512 + o0 + q];
        sA0.y = row[(size_t)(kk + 1) * LD + 512 + o0 + q];
        sA1.x = row[(size_t)(kk + 4) * LD + 512 + o0 + q];
        sA1.y = row[(size_t)(kk + 5) * LD + 512 + o0 + q];

        // B fragments (4 x 16): columns = this wave's i-columns, a2 (256..) / s2 (768..)
        v2f aB0t0, aB1t0, aB0t1, aB1t1, sB0t0, sB1t0, sB0t1, sB1t1;
        aB0t0.x = row[(size_t)(kk + 0) * LD + 256 + c0];
        aB0t0.y = row[(size_t)(kk + 1) * LD + 256 + c0];
        aB1t0.x = row[(size_t)(kk + 4) * LD + 256 + c0];
        aB1t0.y = row[(size_t)(kk + 5) * LD + 256 + c0];
        aB0t1.x = row[(size_t)(kk + 0) * LD + 256 + c1];
        aB0t1.y = row[(size_t)(kk + 1) * LD + 256 + c1];
        aB1t1.x = row[(size_t)(kk + 4) * LD + 256 + c1];
        aB1t1.y = row[(size_t)(kk + 5) * LD + 256 + c1];
        sB0t0.x = row[(size_t)(kk + 0) * LD + 768 + c0];
        sB0t0.y = row[(size_t)(kk + 1) * LD + 768 + c0];
        sB1t0.x = row[(size_t)(kk + 4) * LD + 768 + c0];
        sB1t0.y = row[(size_t)(kk + 5) * LD + 768 + c0];
        sB0t1.x = row[(size_t)(kk + 0) * LD + 768 + c1];
        sB0t1.y = row[(size_t)(kk + 1) * LD + 768 + c1];
        sB1t1.x = row[(size_t)(kk + 4) * LD + 768 + c1];
        sB1t1.y = row[(size_t)(kk + 5) * LD + 768 + c1];

        // add/sub tiles: K=8 -> two chained 16x16x4 WMMAs per tile
        v8f ad0 = __builtin_amdgcn_wmma_f32_16x16x4_f32(false, aA0, false, aB0t0, (short)0, zero, false, false);
        ad0     = __builtin_amdgcn_wmma_f32_16x16x4_f32(false, aA1, false, aB1t0, (short)0, ad0,  false, false);
        v8f ad1 = __builtin_amdgcn_wmma_f32_16x16x4_f32(false, aA0, false, aB0t1, (short)0, zero, false, false);
        ad1     = __builtin_amdgcn_wmma_f32_16x16x4_f32(false, aA1, false, aB1t1, (short)0, ad1,  false, false);
        v8f su0 = __builtin_amdgcn_wmma_f32_16x16x4_f32(false, sA0, false, sB0t0, (short)0, zero, false, false);
        su0     = __builtin_amdgcn_wmma_f32_16x16x4_f32(false, sA1, false, sB1t0, (short)0, su0,  false, false);
        v8f su1 = __builtin_amdgcn_wmma_f32_16x16x4_f32(false, sA0, false, sB0t1, (short)0, zero, false, false);
        su1     = __builtin_amdgcn_wmma_f32_16x16x4_f32(false, sA1, false, sB1t1, (short)0, su1,  false, false);

        // fw += add*sa - sub*ss  (layouts match: reg r -> o = o0 + 8h + r)
        #pragma unroll
        for (int r = 0; r < 8; ++r) {
            fw0[r] = fmaf(ad0[r], sa, fw0[r]);
            fw0[r] = fmaf(su0[r], -ss, fw0[r]);
            fw1[r] = fmaf(ad1[r], sa, fw1[r]);
            fw1[r] = fmaf(su1[r], -ss, fw1[r]);
        }

        // GEMV partials over this lane's two columns
        float xv0 = x[(size_t)(s * BATCH + b) * 256 + c0];
        float xv1 = x[(size_t)(s * BATCH + b) * 256 + c1];
        float pr[8];
        #pragma unroll
        for (int r = 0; r < 8; ++r)
            pr[r] = fw0[r] * xv0 + fw1[r] * xv1;

        // reduce over the 16-lane halves (masks < 16 stay within a half)
        #pragma unroll
        for (int m = 1; m <= 8; m <<= 1) {
            #pragma unroll
            for (int r = 0; r < 8; ++r)
                pr[r] += __shfl_xor(pr[r], m, 32);
        }

        if (q == 0) {
            #pragma unroll
            for (int r = 0; r < 8; ++r)
                red[w * 16 + 8 * h + r] = pr[r];   // half h -> o = 8h + r
        }
        __syncthreads();

        if (tid < 16) {
            float v = rowb[1026 + o0 + tid];       // precomputed w0 @ x base
            #pragma unroll
            for (int ww = 0; ww < 8; ++ww)
                v += red[ww * 16 + tid];
            out[(size_t)(s * BATCH + b) * 256 + o0 + tid] = v;
        }
        __syncthreads();
    }
}

// ---------------------------------------------------------------------------
extern "C" void kernel_launch(void* const* d_in, const int* in_sizes, int n_in,
                              void* d_out, int out_size, void* d_ws, size_t ws_size,
                              hipStream_t stream)
{
    const float* x       = (const float*)d_in[0];   // (512, 8, 256)
    const float* slow_W  = (const float*)d_in[1];   // (1026, 256)
    const float* slow_b  = (const float*)d_in[2];   // (1026,)
    const float* fast_W0 = (const float*)d_in[3];   // (256, 256)
    float*       out     = (float*)d_out;           // (512, 8, 256)
    float*       wu      = (float*)d_ws;            // 4096 x 1282 f32 (~21 MB, L2-resident)

    // Stage 1: fused slow GEMM + w0-base GEMM (WMMA f32)
    int total_waves = MTILES * NTILES;
    int blocks1 = (total_waves + 7) / 8;
    fwp_slow_gemm<<<blocks1, 256, 0, stream>>>(x, slow_W, slow_b, fast_W0, wu);

    // Stage 2: sequential fast-weight scan (register-resident fw, WMMA updates)
    dim3 grid2(16, BATCH);
    fwp_scan<<<grid2, 256, 0, stream>>>(wu, x, out);
}